// LineVectorizer_24189255811231
// MI455X (gfx1250) — compile-verified
//
#include <hip/hip_runtime.h>
#include <hip/hip_bf16.h>
#include <cstdint>
#include <cstddef>

#define BN_EPS 1e-5f

typedef __attribute__((ext_vector_type(2))) float v2f;
typedef __attribute__((ext_vector_type(8))) float v8f;

// =====================================================================
// Generic fp32 GEMM on V_WMMA_F32_16X16X4_F32.
//   C[M,N] = epilogue(A[M,K] @ B[K,N])
// 256 threads = 8 waves. M tile = 64 rows/WG. Wave w: msub = w&3 (16 rows),
// nhalf = w>>2 (N/2 cols -> NT = N/32 16-wide tiles).
// K staged in 64-wide chunks via LDS; A transposed to [k][m] (pitch 72 so
// the two half-waves of a WMMA A-fragment read disjoint bank groups).
//
// AMODE: 0 = A[m,k] at Ab[k*65536 + m]            (BCHW feature, K1)
//        1 = row-major, row stride KDIM           (K3, K5)
//        2 = padded conv rows: base=(m>>3)*640+(m&7)*64, contiguous 192 (K4)
// BMODE: 0 = B[k,n] at Bm[n*KDIM + k]
//        1 = B[k,n] at Bm[n*192 + (k&63)*3 + (k>>6)]   (w_c2 OIH im2col)
// ATRANS: apply bn(at*)+relu per-k while staging A (fuses bn1+relu).
// EPI: 0 bias->NHWC x | 1 bias+bn+relu->padded | 2 bias+bn+relu->dense
//      3 bias+residual+relu (in-place capable)
// =====================================================================
template<int KDIM, int NDIM, int AMODE, int BMODE, int EPI, int ATRANS>
__global__ __launch_bounds__(256)
void gemm_wmma_kernel(const float* __restrict__ A,
                      const float* __restrict__ Bm,
                      float* __restrict__ C,
                      const float* __restrict__ bias,
                      const float* __restrict__ bnG,
                      const float* __restrict__ bnB,
                      const float* __restrict__ bnM,
                      const float* __restrict__ bnV,
                      const float* __restrict__ atG,
                      const float* __restrict__ atB,
                      const float* __restrict__ atM,
                      const float* __restrict__ atV,
                      const float* __restrict__ resid,
                      int aBatchStride, int cBatchStride)
{
    constexpr int KC = 64;           // K chunk through LDS
    constexpr int AP = 72;           // A LDS pitch ([k][m])
    constexpr int BP = NDIM + 8;     // B LDS pitch ([k][n])
    constexpr int NT = NDIM / 32;    // 16-wide n tiles per wave

    __shared__ float As[KC * AP];
    __shared__ float Bs[KC * BP];

    const int tid   = threadIdx.x;
    const int lane  = tid & 31;
    const int wv    = tid >> 5;
    const int msub  = wv & 3;
    const int nhalf = wv >> 2;
    const int m0    = blockIdx.x * 64;

    const float* Ab = A + (size_t)blockIdx.y * (size_t)aBatchStride;
    float*       Cb = C + (size_t)blockIdx.y * (size_t)cBatchStride;

    v8f zero;
    #pragma unroll
    for (int r = 0; r < 8; ++r) zero[r] = 0.0f;
    v8f acc[NT];
    #pragma unroll
    for (int t = 0; t < NT; ++t) acc[t] = zero;

    const int klane = (lane >> 4) << 1;               // 0 (lanes 0-15) / 2
    const int mlane = msub * 16 + (lane & 15);
    const int nlane = nhalf * (NDIM / 2) + (lane & 15);

    for (int k0 = 0; k0 < KDIM; k0 += KC) {
        // ---------- stage A chunk: As[k][m] = A[m0+m][k0+k] ----------
        if constexpr (AMODE == 0) {
            #pragma unroll
            for (int it = 0; it < (KC * 64 / 4) / 256; ++it) {
                int e  = tid + it * 256;
                int k  = e >> 4;
                int m4 = (e & 15) << 2;
                const float4 v =
                    *(const float4*)(Ab + (size_t)(k0 + k) * 65536 + m0 + m4);
                As[k * AP + m4 + 0] = v.x;
                As[k * AP + m4 + 1] = v.y;
                As[k * AP + m4 + 2] = v.z;
                As[k * AP + m4 + 3] = v.w;
            }
        } else {
            #pragma unroll
            for (int it = 0; it < (KC * 64) / 256; ++it) {
                int e  = tid + it * 256;
                int m  = e >> 6;
                int k  = e & 63;
                int mm = m0 + m;
                int rowBase;
                if constexpr (AMODE == 2)
                    rowBase = (mm >> 3) * 640 + (mm & 7) * 64;
                else
                    rowBase = mm * KDIM;
                float v = Ab[(size_t)rowBase + k0 + k];
                if constexpr (ATRANS) {
                    int kk = k0 + k;
                    v = fmaxf((v - atM[kk]) * (atG[kk] * rsqrtf(atV[kk] + BN_EPS))
                                  + atB[kk], 0.0f);
                }
                As[k * AP + m] = v;
            }
        }
        // ---------- stage B chunk: Bs[k][n] ----------
        #pragma unroll
        for (int it = 0; it < (KC * NDIM) / 256; ++it) {
            int e  = tid + it * 256;
            int n  = e >> 6;
            int k  = e & 63;
            int kk = k0 + k;
            float v;
            if constexpr (BMODE == 0)
                v = Bm[n * KDIM + kk];
            else
                v = Bm[n * 192 + (kk & 63) * 3 + (kk >> 6)];
            Bs[k * BP + n] = v;
        }
        __syncthreads();
        // ---------- 16 WMMA k-steps over this chunk ----------
        #pragma unroll
        for (int kk = 0; kk < KC; kk += 4) {
            v2f a;
            a.x = As[(kk + klane)     * AP + mlane];
            a.y = As[(kk + klane + 1) * AP + mlane];
            #pragma unroll
            for (int t = 0; t < NT; ++t) {
                v2f b;
                b.x = Bs[(kk + klane)     * BP + nlane + t * 16];
                b.y = Bs[(kk + klane + 1) * BP + nlane + t * 16];
                acc[t] = __builtin_amdgcn_wmma_f32_16x16x4_f32(
                    false, a, false, b, (short)0, acc[t], false, false);
            }
        }
        __syncthreads();
    }

    // ---------- epilogue / store (D layout: N = lane&15, M = r + 8*(lane>=16))
    #pragma unroll
    for (int t = 0; t < NT; ++t) {
        int nn = nlane + t * 16;
        #pragma unroll
        for (int r = 0; r < 8; ++r) {
            int mm  = m0 + msub * 16 + ((lane >> 4) << 3) + r;
            float v = acc[t][r];
            if constexpr (EPI == 0) {
                v += bias[nn];
                Cb[(size_t)mm * NDIM + nn] = v;
            } else if constexpr (EPI == 1) {
                v += bias[nn];
                v = fmaxf((v - bnM[nn]) * (bnG[nn] * rsqrtf(bnV[nn] + BN_EPS))
                              + bnB[nn], 0.0f);
                Cb[(size_t)(mm >> 3) * 640 + (mm & 7) * 64 + 64 + nn] = v;
            } else if constexpr (EPI == 2) {
                v += bias[nn];
                v = fmaxf((v - bnM[nn]) * (bnG[nn] * rsqrtf(bnV[nn] + BN_EPS))
                              + bnB[nn], 0.0f);
                Cb[(size_t)mm * 64 + nn] = v;
            } else {
                v += bias[nn] + resid[(size_t)mm * 128 + nn];
                Cb[(size_t)mm * 128 + nn] = fmaxf(v, 0.0f);
            }
        }
    }
}

// =====================================================================
// K2: line point sampling (bilinear on NHWC x, L2-resident) + maxpool(4)
// One block (128 threads = channels) per line. Also zeroes the conv-pad
// rows of the padded H2 buffer so K4's im2col is contiguous reads.
// =====================================================================
__global__ __launch_bounds__(128)
void sample_pool_kernel(const float* __restrict__ X,     // (B,256,256,128)
                        const float* __restrict__ lines, // (B,nl,2,2)
                        float* __restrict__ XPml,        // (B*nl*8, 128)
                        float* __restrict__ H2pad,       // (B*nl, 10, 64)
                        int nl)
{
    const int n = blockIdx.x;
    const int c = threadIdx.x;
    const int b = n / nl;
    const float* lp = lines + (size_t)n * 4;
    const float ax = lp[0], ay = lp[1], ex = lp[2], ey = lp[3];
    const float* Xb = X + (size_t)b * 256 * 256 * 128;

    float pool[8];
    #pragma unroll
    for (int l = 0; l < 8; ++l) pool[l] = -3.402823466e38f;

    for (int p = 0; p < 32; ++p) {
        float lam = (float)p * (1.0f / 31.0f);
        float px = ax * lam + ex * (1.0f - lam) - 0.5f;
        float py = ay * lam + ey * (1.0f - lam) - 0.5f;
        float px0 = fminf(fmaxf(floorf(px), 0.0f), 255.0f);
        float py0 = fminf(fmaxf(floorf(py), 0.0f), 255.0f);
        float px1 = fminf(px0 + 1.0f, 255.0f);
        float py1 = fminf(py0 + 1.0f, 255.0f);
        int i0 = (int)px0, j0 = (int)py0, i1 = (int)px1, j1 = (int)py1;
        float w00 = (px1 - px) * (py1 - py);
        float w10 = (px - px0) * (py1 - py);
        float w01 = (px1 - px) * (py - py0);
        float w11 = (px - px0) * (py - py0);
        float v = w00 * Xb[(size_t)((i0 << 8) + j0) * 128 + c]
                + w10 * Xb[(size_t)((i1 << 8) + j0) * 128 + c]
                + w01 * Xb[(size_t)((i0 << 8) + j1) * 128 + c]
                + w11 * Xb[(size_t)((i1 << 8) + j1) * 128 + c];
        pool[p >> 2] = fmaxf(pool[p >> 2], v);
    }
    #pragma unroll
    for (int l = 0; l < 8; ++l)
        XPml[((size_t)n * 8 + l) * 128 + c] = pool[l];

    if (c < 64) {
        H2pad[(size_t)n * 640 + c]          = 0.0f;   // pad row l=-1
        H2pad[(size_t)n * 640 + 9 * 64 + c] = 0.0f;   // pad row l=8
    }
}

// =====================================================================
// K6: out[n,j] = fc2_b[j] + sum_k relu'd_feat[n,k] * fc2_w[j,k]
// feat[n, c*8+l] = XPF[(n*8+l)*128 + c]. One wave32 per line.
// =====================================================================
__global__ __launch_bounds__(256)
void fc2_kernel(const float* __restrict__ XPF, const float* __restrict__ W,
                const float* __restrict__ bvec, float* __restrict__ out, int Nl)
{
    const int lane = threadIdx.x & 31;
    const int n = blockIdx.x * 8 + (threadIdx.x >> 5);
    if (n >= Nl) return;
    const float* row = XPF + (size_t)n * 8 * 128;
    float a0 = 0.f, a1 = 0.f, a2 = 0.f, a3 = 0.f;
    for (int k = lane; k < 1024; k += 32) {
        int cch = k >> 3, l = k & 7;
        float v = row[l * 128 + cch];
        a0 += v * W[k];
        a1 += v * W[1024 + k];
        a2 += v * W[2048 + k];
        a3 += v * W[3072 + k];
    }
    #pragma unroll
    for (int off = 16; off; off >>= 1) {
        a0 += __shfl_xor(a0, off, 32);
        a1 += __shfl_xor(a1, off, 32);
        a2 += __shfl_xor(a2, off, 32);
        a3 += __shfl_xor(a3, off, 32);
    }
    if (lane == 0) {
        out[(size_t)n * 4 + 0] = a0 + bvec[0];
        out[(size_t)n * 4 + 1] = a1 + bvec[1];
        out[(size_t)n * 4 + 2] = a2 + bvec[2];
        out[(size_t)n * 4 + 3] = a3 + bvec[3];
    }
}

extern "C" void kernel_launch(void* const* d_in, const int* in_sizes, int n_in,
                              void* d_out, int out_size, void* d_ws, size_t ws_size,
                              hipStream_t stream)
{
    const float* feature = (const float*)d_in[0];
    const float* lines   = (const float*)d_in[1];
    const float* w_fc1   = (const float*)d_in[2];
    const float* b_fc1   = (const float*)d_in[3];
    const float* bn1_g   = (const float*)d_in[4];
    const float* bn1_b   = (const float*)d_in[5];
    const float* bn1_m   = (const float*)d_in[6];
    const float* bn1_v   = (const float*)d_in[7];
    const float* w_c1    = (const float*)d_in[8];
    const float* b_c1    = (const float*)d_in[9];
    const float* bn2_g   = (const float*)d_in[10];
    const float* bn2_b   = (const float*)d_in[11];
    const float* bn2_m   = (const float*)d_in[12];
    const float* bn2_v   = (const float*)d_in[13];
    const float* w_c2    = (const float*)d_in[14];
    const float* b_c2    = (const float*)d_in[15];
    const float* bn3_g   = (const float*)d_in[16];
    const float* bn3_b   = (const float*)d_in[17];
    const float* bn3_m   = (const float*)d_in[18];
    const float* bn3_v   = (const float*)d_in[19];
    const float* w_c3    = (const float*)d_in[20];
    const float* b_c3    = (const float*)d_in[21];
    const float* fc2_w   = (const float*)d_in[22];
    const float* fc2_b   = (const float*)d_in[23];

    const int B  = in_sizes[0] / (256 * 256 * 256);   // 2
    const int nl = in_sizes[1] / (B * 4);             // 4000
    const int NL = B * nl;                            // 8000
    const int M  = NL * 8;                            // 64000

    float* ws = (float*)d_ws;
    float* X     = ws;                      size_t off = (size_t)B * 256 * 256 * 128;
    float* XPml  = ws + off;                off += (size_t)M * 128;
    float* H2pad = ws + off;                off += (size_t)NL * 640;
    float* H3    = ws + off;                // + M*64

    // K1: x = conv1x1(feature, w_fc1)+b  ->  NHWC (B,256,256,128)
    gemm_wmma_kernel<256, 128, 0, 0, 0, 0>
        <<<dim3(65536 / 64, B), 256, 0, stream>>>(
            feature, w_fc1, X, b_fc1,
            nullptr, nullptr, nullptr, nullptr,
            nullptr, nullptr, nullptr, nullptr,
            nullptr, 256 * 65536, 65536 * 128);

    // K2: bilinear sample 32 pts/line + maxpool(4) -> XPml (m=(n,l), c)
    sample_pool_kernel<<<NL, 128, 0, stream>>>(X, lines, XPml, H2pad, nl);

    // K3: h2 = relu(bn2(relu(bn1(xp)) @ w_c1 + b_c1)) -> padded (n,10,64)
    //      (bn1+relu fused into A staging)
    gemm_wmma_kernel<128, 64, 1, 0, 1, 1>
        <<<dim3(M / 64, 1), 256, 0, stream>>>(
            XPml, w_c1, H2pad, b_c1,
            bn2_g, bn2_b, bn2_m, bn2_v,
            bn1_g, bn1_b, bn1_m, bn1_v,
            nullptr, 0, 0);

    // K4: h3 = relu(bn3(conv1d_k3(h2) + b_c2)) as GEMM K=192 over padded rows
    gemm_wmma_kernel<192, 64, 2, 1, 2, 0>
        <<<dim3(M / 64, 1), 256, 0, stream>>>(
            H2pad, w_c2, H3, b_c2,
            bn3_g, bn3_b, bn3_m, bn3_v,
            nullptr, nullptr, nullptr, nullptr,
            nullptr, 0, 0);

    // K5: xpf = relu(xp + h3 @ w_c3 + b_c3), in place into XPml
    gemm_wmma_kernel<64, 128, 1, 0, 3, 0>
        <<<dim3(M / 64, 1), 256, 0, stream>>>(
            H3, w_c3, XPml, b_c3,
            nullptr, nullptr, nullptr, nullptr,
            nullptr, nullptr, nullptr, nullptr,
            XPml, 0, 0);

    // K6: out = feat @ fc2_w^T + fc2_b
    fc2_kernel<<<NL / 8, 256, 0, stream>>>(XPml, fc2_w, fc2_b, (float*)d_out, NL);
}